// AttentionFusion4_51127290692314
// MI455X (gfx1250) — compile-verified
//
#include <hip/hip_runtime.h>
#include <cstdint>

#define EPS 1e-8f

typedef float v4f __attribute__((ext_vector_type(4)));

// One block per row. 256 threads * float4 = 1024 floats = one full row of A and B.
// A/B rows are staged global->LDS with CDNA5 async-to-LDS (ASYNCcnt), then each
// thread consumes exactly the bytes it staged (no barrier needed before use).
__global__ __launch_bounds__(256)
void cosmerge_kernel(const float* __restrict__ A,
                     const float* __restrict__ B,
                     float* __restrict__ out,
                     int D /* = 1024 */)
{
    __shared__ v4f  sA[256];
    __shared__ v4f  sB[256];
    __shared__ float red[3][8];

    const int t   = threadIdx.x;
    const int row = blockIdx.x;

    // Byte offset of this thread's float4 within the [N,D] matrix.
    // Max = 16384*4096 B = 64 MiB, fits the signed 32-bit VADDR of GVS mode.
    const uint32_t goff = (uint32_t)row * ((uint32_t)D * 4u) + (uint32_t)t * 16u;

    // Low 32 bits of a generic pointer to __shared__ == wave-relative LDS byte
    // address (flat aperture keeps the LDS offset in addr[31:0]).
    const uint32_t ldsA = (uint32_t)(uintptr_t)&sA[t];
    const uint32_t ldsB = (uint32_t)(uintptr_t)&sB[t];

    const uint64_t baseA = (uint64_t)(uintptr_t)A;
    const uint64_t baseB = (uint64_t)(uintptr_t)B;

    // CDNA5 async global->LDS, GVS addressing: mem = SADDR + VADDR, LDS = VDST.
    asm volatile("global_load_async_to_lds_b128 %0, %1, %2"
                 :: "v"(ldsA), "v"(goff), "s"(baseA) : "memory");
    asm volatile("global_load_async_to_lds_b128 %0, %1, %2"
                 :: "v"(ldsB), "v"(goff), "s"(baseB) : "memory");
    asm volatile("s_wait_asynccnt 0" ::: "memory");

    const v4f a = sA[t];
    const v4f b = sB[t];

    // Per-thread partials for the three row reductions.
    float dot = a.x * b.x + a.y * b.y + a.z * b.z + a.w * b.w;
    float na2 = a.x * a.x + a.y * a.y + a.z * a.z + a.w * a.w;
    float nb2 = b.x * b.x + b.y * b.y + b.z * b.z + b.w * b.w;

    // wave32 butterfly reduction.
    #pragma unroll
    for (int off = 16; off > 0; off >>= 1) {
        dot += __shfl_xor(dot, off, 32);
        na2 += __shfl_xor(na2, off, 32);
        nb2 += __shfl_xor(nb2, off, 32);
    }

    const int wave = t >> 5;
    if ((t & 31) == 0) {
        red[0][wave] = dot;
        red[1][wave] = na2;
        red[2][wave] = nb2;
    }
    __syncthreads();

    // Every thread folds the 8 wave partials (cheap; avoids a 2nd barrier).
    float dS = 0.f, aS = 0.f, bS = 0.f;
    #pragma unroll
    for (int i = 0; i < 8; ++i) {
        dS += red[0][i];
        aS += red[1][i];
        bS += red[2][i];
    }

    const float scale = fmaxf(sqrtf(aS) * sqrtf(bS), EPS);
    const float w = 1.0f - dS / scale;     // cosine distance
    const float u = 2.0f - w;

    v4f m;
    m.x = 0.5f * (w * a.x + u * b.x);
    m.y = 0.5f * (w * a.y + u * b.y);
    m.z = 0.5f * (w * a.z + u * b.z);
    m.w = 0.5f * (w * a.w + u * b.w);

    // Streaming output: 128-bit non-temporal store (never re-read).
    v4f* orow = reinterpret_cast<v4f*>(out + (size_t)row * (size_t)D);
    __builtin_nontemporal_store(m, orow + t);
}

extern "C" void kernel_launch(void* const* d_in, const int* in_sizes, int n_in,
                              void* d_out, int out_size, void* d_ws, size_t ws_size,
                              hipStream_t stream) {
    (void)n_in; (void)out_size; (void)d_ws; (void)ws_size;
    const float* A = (const float*)d_in[0];
    const float* B = (const float*)d_in[1];
    float* out = (float*)d_out;

    const int D = 1024;
    const int N = in_sizes[0] / D;   // 16384

    cosmerge_kernel<<<N, 256, 0, stream>>>(A, B, out, D);
}